// LiZAttention_3040836846396
// MI455X (gfx1250) — compile-verified
//
#include <hip/hip_runtime.h>
#include <cstdint>

// ---------------------------------------------------------------------------
// LiZAttention for MI455X (gfx1250): fp32 WMMA (v_wmma_f32_16x16x4_f32) for all
// GEMM-shaped work, TDM (tensor_load_to_lds) for attention tile staging,
// wave32 shuffle reductions, register-resident delta-rule scan.
// ---------------------------------------------------------------------------

typedef float v2f __attribute__((ext_vector_type(2)));
typedef float v8f __attribute__((ext_vector_type(8)));

#define NSEQ 2048
#define DMODEL 1024
#define NHEAD 16
#define NKV 8
#define HD 64

__device__ __forceinline__ v8f wmma4(v2f a, v2f b, v8f c) {
  // D = A(16x4) * B(4x16) + C(16x16), fp32
  return __builtin_amdgcn_wmma_f32_16x16x4_f32(
      /*neg_a=*/false, a, /*neg_b=*/false, b,
      /*c_mod=*/(short)0, c, /*reuse_a=*/false, /*reuse_b=*/false);
}

// Fragment fetch: for the 32-bit A(16x4)/B(4x16) lane layouts where the 16-wide
// dimension is the LDS row, lane L wants the contiguous float pair at
// [row = L%16][col = 4*c + 2*(L/16)].
__device__ __forceinline__ v2f lds_frag(const float* base, int row, int stride,
                                        int c, int hi) {
  const float* p = base + row * stride + 4 * c + 2 * hi;
  v2f r;
  r.x = p[0];
  r.y = p[1];
  return r;
}

// B-operand fragment when the K-dim of B advances along the LDS *row* index
// (row-major [k][n] tile): b.x = base[(4c+2hi)*stride + n], b.y one row below.
__device__ __forceinline__ v2f lds_frag_rowpair(const float* base, int n,
                                                int stride, int c, int hi) {
  const float* p = base + (4 * c + 2 * hi) * stride + n;
  v2f r;
  r.x = p[0];
  r.y = p[stride];
  return r;
}

// ---------------------------------------------------------------------------
// Tensor Data Mover: 2D tile load Global -> LDS (CDNA5 D#, ISA 08 §8.3-8.5).
// Toolchain probe: clang-23 lane exposes the 6-arg builtin
//   (uint32x4 g0, int32x8 g1, int32x4 g2, int32x4 g3, int32x8 extra, i32 cpol)
// ---------------------------------------------------------------------------
#if defined(__HIP_DEVICE_COMPILE__) && __has_builtin(__builtin_amdgcn_tensor_load_to_lds)
#define HAVE_TDM 1
typedef unsigned int u32x4 __attribute__((ext_vector_type(4)));
typedef int i32x4 __attribute__((ext_vector_type(4)));
typedef int i32x8 __attribute__((ext_vector_type(8)));

__device__ __forceinline__ void tdm_load_2d(unsigned lds_addr, const float* gptr,
                                            int tile_w, int tile_h,
                                            int row_stride, int tdim0, int tdim1) {
  unsigned long long ga = (unsigned long long)(uintptr_t)gptr;
  u32x4 g0;
  g0.x = 1u;                                   // count=1 user descriptor
  g0.y = lds_addr;                             // [63:32] lds_addr
  g0.z = (unsigned)(ga & 0xffffffffu);         // [95:64] global_addr lo
  g0.w = (unsigned)((ga >> 32) & 0x01ffffffu)  // [120:96] global_addr hi
         | (2u << 30);                         // [127:126] type=2 (image)
  i32x8 g1;
  g1[0] = (2 << 16);                                   // data_size=4B, mask=0
  g1[1] = (int)(((unsigned)tdim0 & 0xffffu) << 16);    // tensor_dim0[15:0]
  g1[2] = (int)((((unsigned)tdim0 >> 16) & 0xffffu) |
                (((unsigned)tdim1 & 0xffffu) << 16));  // dim0 hi | dim1 lo
  g1[3] = (int)((((unsigned)tdim1 >> 16) & 0xffffu) |
                ((unsigned)tile_w << 16));             // dim1 hi | tile_dim0
  g1[4] = tile_h;                                      // tile_dim1 (tile_dim2=0)
  g1[5] = row_stride;                                  // tensor_dim0_stride lo
  g1[6] = 0;
  g1[7] = 0;
  i32x4 gz4 = {0, 0, 0, 0};
  i32x8 gz8 = {0, 0, 0, 0, 0, 0, 0, 0};
  __builtin_amdgcn_tensor_load_to_lds(g0, g1, gz4, gz4, gz8, 0);
}
#else
#define HAVE_TDM 0
#endif

// ---------------------------------------------------------------------------
// fp32 WMMA GEMM: C[M,N] = A[M,K] @ B[K,N].  M % 128 == 0, N % 64 == 0,
// K % 16 == 0.  128 threads = 4 waves; wave w owns 32 rows (2 A-frag rows,
// 8 C-tiles) so each B-fragment feeds two WMMAs.
// ---------------------------------------------------------------------------
__global__ __launch_bounds__(128) void gemm_f32(const float* __restrict__ A,
                                                const float* __restrict__ B,
                                                float* __restrict__ C,
                                                int M, int N, int K) {
  __shared__ float sA[128 * 16];  // [m][k], stride 16
  __shared__ float sBt[64 * 16];  // [n][k], stride 16 (B transposed)

  const int tid = threadIdx.x;
  const int lane = tid & 31;
  const int wv = tid >> 5;
  const int hi = lane >> 4;
  const int ln = lane & 15;
  const int m0 = blockIdx.y * 128;
  const int n0 = blockIdx.x * 64;

  v8f vzero = {0, 0, 0, 0, 0, 0, 0, 0};
  v8f acc[2][4];
#pragma unroll
  for (int u = 0; u < 2; ++u)
#pragma unroll
    for (int t = 0; t < 4; ++t) acc[u][t] = vzero;

  for (int k0 = 0; k0 < K; k0 += 16) {
    __syncthreads();
    // Stage A tile [128x16]
#pragma unroll
    for (int i = 0; i < 16; ++i) {
      int e = tid + i * 128;  // 0..2047
      int r = e >> 4, c = e & 15;
      sA[e] = A[(size_t)(m0 + r) * K + k0 + c];
    }
    // Stage B tile transposed: sBt[n][c] = B[k0+c][n0+n]
#pragma unroll
    for (int i = 0; i < 8; ++i) {
      int e = tid + i * 128;
      int c = e >> 6, n = e & 63;
      sBt[n * 16 + c] = B[(size_t)(k0 + c) * N + n0 + n];
    }
    if (k0 + 16 < K) {  // hint next tiles into cache (global_prefetch_b8)
      __builtin_prefetch(A + (size_t)(m0 + (tid >> 4)) * K + k0 + 16 + (tid & 15), 0, 3);
      __builtin_prefetch(B + (size_t)(k0 + 16 + (tid >> 6)) * N + n0 + (tid & 63), 0, 3);
    }
    __syncthreads();
#pragma unroll
    for (int kc = 0; kc < 4; ++kc) {
      v2f a0 = lds_frag(sA, wv * 32 + ln, 16, kc, hi);
      v2f a1 = lds_frag(sA, wv * 32 + 16 + ln, 16, kc, hi);
#pragma unroll
      for (int t = 0; t < 4; ++t) {
        v2f b = lds_frag(sBt, t * 16 + ln, 16, kc, hi);
        acc[0][t] = wmma4(a0, b, acc[0][t]);
        acc[1][t] = wmma4(a1, b, acc[1][t]);
      }
    }
  }

#pragma unroll
  for (int u = 0; u < 2; ++u)
#pragma unroll
    for (int t = 0; t < 4; ++t)
#pragma unroll
      for (int r = 0; r < 8; ++r) {
        int row = m0 + wv * 32 + u * 16 + r + 8 * hi;  // C: M = reg + 8*(lane/16)
        int col = n0 + t * 16 + ln;
        C[(size_t)row * N + col] = acc[u][t][r];
      }
}

// ---------------------------------------------------------------------------
// Row transform: softmax over 64-wide rows, clipped to [eps, 1-eps]; for k
// rows also gl = clip(log_sigmoid(raw)/16, ±16).  One wave per row.
// ---------------------------------------------------------------------------
__global__ __launch_bounds__(128) void row_transform(const float* __restrict__ X,
                                                     float* __restrict__ SM,
                                                     float* __restrict__ GL,
                                                     int nrows) {
  const int lane = threadIdx.x & 31;
  const int row = blockIdx.x * 4 + (threadIdx.x >> 5);
  if (row >= nrows) return;
  const float* x = X + (size_t)row * 64;
  float x0 = x[lane];
  float x1 = x[lane + 32];

  float mx = fmaxf(x0, x1);
  mx = fmaxf(mx, __shfl_xor(mx, 1));
  mx = fmaxf(mx, __shfl_xor(mx, 2));
  mx = fmaxf(mx, __shfl_xor(mx, 4));
  mx = fmaxf(mx, __shfl_xor(mx, 8));
  mx = fmaxf(mx, __shfl_xor(mx, 16));

  float e0 = __expf(x0 - mx), e1 = __expf(x1 - mx);
  float s = e0 + e1;
  s += __shfl_xor(s, 1);
  s += __shfl_xor(s, 2);
  s += __shfl_xor(s, 4);
  s += __shfl_xor(s, 8);
  s += __shfl_xor(s, 16);
  float inv = 1.0f / s;

  const float EPS = 1e-6f;
  SM[(size_t)row * 64 + lane] = fminf(fmaxf(e0 * inv, EPS), 1.0f - EPS);
  SM[(size_t)row * 64 + lane + 32] = fminf(fmaxf(e1 * inv, EPS), 1.0f - EPS);

  if (GL != nullptr) {
    float g0 = (fminf(x0, 0.0f) - log1pf(__expf(-fabsf(x0)))) * (1.0f / 16.0f);
    float g1 = (fminf(x1, 0.0f) - log1pf(__expf(-fabsf(x1)))) * (1.0f / 16.0f);
    GL[(size_t)row * 64 + lane] = fminf(fmaxf(g0, -16.0f), 16.0f);
    GL[(size_t)row * 64 + lane + 32] = fminf(fmaxf(g1, -16.0f), 16.0f);
  }
}

// ---------------------------------------------------------------------------
// Gated delta-rule scan.  One workgroup (64 threads) per head; thread t owns
// state column m=t (S[d][t] in registers).
// ---------------------------------------------------------------------------
__global__ __launch_bounds__(64) void delta_rule(const float* __restrict__ ql,
                                                 const float* __restrict__ kl,
                                                 const float* __restrict__ v,
                                                 const float* __restrict__ gl,
                                                 float* __restrict__ olin) {
  const int h = blockIdx.x;
  const int kvh = h >> 1;  // G = 2
  const int t = threadIdx.x;

  __shared__ float sq[64], sk[64], seg[64];

  float S[64];
#pragma unroll
  for (int d = 0; d < 64; ++d) S[d] = 0.0f;

  for (int n = 0; n < NSEQ; ++n) {
    float qv = ql[(size_t)n * DMODEL + h * HD + t];
    float kv = kl[(size_t)n * (NKV * HD) + kvh * HD + t];
    float vv = v[(size_t)n * (NKV * HD) + kvh * HD + t];
    float gv = gl[(size_t)n * (NKV * HD) + kvh * HD + t];
    sq[t] = qv;
    sk[t] = kv;
    seg[t] = __expf(gv);
    __syncthreads();

    float err = vv;
#pragma unroll
    for (int d = 0; d < 64; ++d) {
      float sd = S[d] * seg[d];
      S[d] = sd;
      err -= sk[d] * sd;
    }
    float o = 0.0f;
#pragma unroll
    for (int d = 0; d < 64; ++d) {
      float sd = S[d] + sk[d] * err;
      S[d] = sd;
      o += sq[d] * sd;
    }
    olin[(size_t)n * DMODEL + h * HD + t] = o;
    __syncthreads();
  }
}

// ---------------------------------------------------------------------------
// Base causal softmax attention, flash style, fp32 WMMA.  K/V tiles (32x64)
// staged into LDS by the Tensor Data Mover (one wave issues, TENSORcnt wait),
// with a cooperative-load fallback if the TDM builtin is unavailable.
// ---------------------------------------------------------------------------
__global__ __launch_bounds__(128) void attn_base(const float* __restrict__ Q,
                                                 const float* __restrict__ K,
                                                 const float* __restrict__ V,
                                                 float* __restrict__ Ob) {
  const int qt = blockIdx.x;
  const int h = blockIdx.y;
  const int kvh = h >> 1;
  const int tid = threadIdx.x;
  const int lane = tid & 31;
  const int wv = tid >> 5;
  const int hi = lane >> 4;
  const int ln = lane & 15;

  __shared__ float sK[32 * 64];     // [key][hd]
  __shared__ float sV[32 * 64];     // [key][hd] (row-major; TDM cannot transpose)
  __shared__ float sP[4][16 * 32];  // per-wave P tile [m][key]

  // Preload Q fragments (A layout, 16 HD-chunks of 4) for this wave's 16 rows
  v2f qa[16];
  {
    const int qrow = qt * 64 + wv * 16 + ln;
    const float* qp = Q + (size_t)qrow * DMODEL + h * HD;
#pragma unroll
    for (int c = 0; c < 16; ++c) {
      qa[c].x = qp[4 * c + 2 * hi];
      qa[c].y = qp[4 * c + 2 * hi + 1];
    }
  }

  v8f vzero = {0, 0, 0, 0, 0, 0, 0, 0};
  v8f O[4];
  O[0] = vzero; O[1] = vzero; O[2] = vzero; O[3] = vzero;
  float mi[8], li[8];
#pragma unroll
  for (int r = 0; r < 8; ++r) { mi[r] = -1e30f; li[r] = 0.0f; }

#if HAVE_TDM
  const unsigned sK_lds = (unsigned)(uintptr_t)(void*)&sK[0];
  const unsigned sV_lds = (unsigned)(uintptr_t)(void*)&sV[0];
#endif

  const float scale = 0.125f;  // 1/sqrt(64)
  const int nkt = 2 * qt + 2;  // causal bound

  for (int kt = 0; kt < nkt; ++kt) {
    const int kbase = kt * 32;
    __syncthreads();  // previous iteration's LDS consumers done
#if HAVE_TDM
    if (wv == 0) {
      const float* kp = K + (size_t)kbase * (NKV * HD) + kvh * HD;
      const float* vp = V + (size_t)kbase * (NKV * HD) + kvh * HD;
      tdm_load_2d(sK_lds, kp, HD, 32, NKV * HD, NKV * HD, NSEQ);
      tdm_load_2d(sV_lds, vp, HD, 32, NKV * HD, NKV * HD, NSEQ);
      __builtin_amdgcn_s_wait_tensorcnt(0);
    }
#else
#pragma unroll
    for (int i = 0; i < 16; ++i) {
      int e = tid + i * 128;  // 0..2047
      int key = e >> 6, hd = e & 63;
      size_t g = (size_t)(kbase + key) * (NKV * HD) + kvh * HD + hd;
      sK[key * 64 + hd] = K[g];
      sV[key * 64 + hd] = V[g];
    }
#endif
    __syncthreads();

    // ---- scores: S = Q @ K^T (two 16x16 tiles per wave) ----
    v8f s0 = vzero, s1 = vzero;
#pragma unroll
    for (int c = 0; c < 16; ++c) {
      v2f b0 = lds_frag(sK, ln, 64, c, hi);       // keys kbase+0..15
      v2f b1 = lds_frag(sK, 16 + ln, 64, c, hi);  // keys kbase+16..31
      s0 = wmma4(qa[c], b0, s0);
      s1 = wmma4(qa[c], b1, s1);
    }

    // ---- scale + causal mask (C layout: row = r + 8*hi, col = lane%16) ----
#pragma unroll
    for (int r = 0; r < 8; ++r) {
      int qg = qt * 64 + wv * 16 + r + 8 * hi;
      int kg0 = kbase + ln;
      int kg1 = kbase + 16 + ln;
      s0[r] = (kg0 <= qg) ? s0[r] * scale : -1e30f;
      s1[r] = (kg1 <= qg) ? s1[r] * scale : -1e30f;
    }

    // ---- online softmax update ----
#pragma unroll
    for (int r = 0; r < 8; ++r) {
      float mx = fmaxf(s0[r], s1[r]);
      mx = fmaxf(mx, __shfl_xor(mx, 1));
      mx = fmaxf(mx, __shfl_xor(mx, 2));
      mx = fmaxf(mx, __shfl_xor(mx, 4));
      mx = fmaxf(mx, __shfl_xor(mx, 8));
      float mn = fmaxf(mi[r], mx);
      float alpha = __expf(mi[r] - mn);
      float p0 = __expf(s0[r] - mn);
      float p1 = __expf(s1[r] - mn);
      s0[r] = p0;
      s1[r] = p1;
      float rs = p0 + p1;
      rs += __shfl_xor(rs, 1);
      rs += __shfl_xor(rs, 2);
      rs += __shfl_xor(rs, 4);
      rs += __shfl_xor(rs, 8);
      li[r] = li[r] * alpha + rs;
      mi[r] = mn;
#pragma unroll
      for (int t = 0; t < 4; ++t) O[t][r] *= alpha;
    }

    // ---- stage P through LDS to re-enter the A-operand layout ----
    float* p = sP[wv];
#pragma unroll
    for (int r = 0; r < 8; ++r) {
      int row = r + 8 * hi;
      p[row * 32 + ln] = s0[r];
      p[row * 32 + 16 + ln] = s1[r];
    }
    // same-wave DS ops are in-order: safe to read back below

    // ---- O += P @ V (8 key-chunks of 4, 4 output hd-tiles) ----
#pragma unroll
    for (int c = 0; c < 8; ++c) {
      v2f a = lds_frag(p, ln, 32, c, hi);
#pragma unroll
      for (int t = 0; t < 4; ++t) {
        v2f b = lds_frag_rowpair(sV, t * 16 + ln, 64, c, hi);
        O[t] = wmma4(a, b, O[t]);
      }
    }
  }

  // ---- normalize and write o_base [N, H*HD] ----
#pragma unroll
  for (int r = 0; r < 8; ++r) {
    float inv = 1.0f / li[r];
    int qg = qt * 64 + wv * 16 + r + 8 * hi;
#pragma unroll
    for (int t = 0; t < 4; ++t) {
      Ob[(size_t)qg * DMODEL + h * HD + t * 16 + ln] = O[t][r] * inv;
    }
  }
}

// ---------------------------------------------------------------------------
__global__ __launch_bounds__(256) void mix_kernel(const float* __restrict__ a,
                                                  const float* __restrict__ b,
                                                  float* __restrict__ out,
                                                  int n) {
  int i = blockIdx.x * 256 + threadIdx.x;
  if (i < n) out[i] = 0.5f * a[i] + 0.5f * b[i];
}

// ---------------------------------------------------------------------------
extern "C" void kernel_launch(void* const* d_in, const int* in_sizes, int n_in,
                              void* d_out, int out_size, void* d_ws,
                              size_t ws_size, hipStream_t stream) {
  (void)in_sizes; (void)n_in; (void)out_size; (void)ws_size;
  const float* hs = (const float*)d_in[0];  // [N, D]
  const float* Wq = (const float*)d_in[1];  // [D, H*HD]
  const float* Wk = (const float*)d_in[2];  // [D, KV*HD]
  const float* Wv = (const float*)d_in[3];  // [D, KV*HD]
  const float* Wo = (const float*)d_in[4];  // [H*HD, D]
  float* out = (float*)d_out;               // [N, D]

  float* ws = (float*)d_ws;
  const size_t NQ = (size_t)NSEQ * DMODEL;
  const size_t NKVE = (size_t)NSEQ * NKV * HD;
  float* q = ws;
  float* k = q + NQ;
  float* v = k + NKVE;
  float* ql = v + NKVE;
  float* kl = ql + NQ;
  float* gl = kl + NKVE;
  float* olin = gl + NKVE;
  float* obase = olin + NQ;
  float* mixb = obase + NQ;

  dim3 blk(128);
  // Projections (fp32 WMMA GEMMs, BM=128 register blocking)
  gemm_f32<<<dim3(DMODEL / 64, NSEQ / 128), blk, 0, stream>>>(hs, Wq, q, NSEQ, DMODEL, DMODEL);
  gemm_f32<<<dim3((NKV * HD) / 64, NSEQ / 128), blk, 0, stream>>>(hs, Wk, k, NSEQ, NKV * HD, DMODEL);
  gemm_f32<<<dim3((NKV * HD) / 64, NSEQ / 128), blk, 0, stream>>>(hs, Wv, v, NSEQ, NKV * HD, DMODEL);

  // Linear-branch feature maps
  row_transform<<<(NSEQ * NHEAD + 3) / 4, 128, 0, stream>>>(q, ql, nullptr, NSEQ * NHEAD);
  row_transform<<<(NSEQ * NKV + 3) / 4, 128, 0, stream>>>(k, kl, gl, NSEQ * NKV);

  // Gated delta-rule scan
  delta_rule<<<16, 64, 0, stream>>>(ql, kl, v, gl, olin);

  // Base causal attention (flash-style WMMA + TDM tile staging)
  attn_base<<<dim3(NSEQ / 64, NHEAD), blk, 0, stream>>>(q, k, v, obase);

  // Mix branches, then final output projection
  mix_kernel<<<(int)(NQ / 256), 256, 0, stream>>>(olin, obase, mixb, (int)NQ);
  gemm_f32<<<dim3(DMODEL / 64, NSEQ / 128), blk, 0, stream>>>(mixb, Wo, out, NSEQ, DMODEL, DMODEL);
}